// FilterDetections_18906446037164
// MI455X (gfx1250) — compile-verified
//
#include <hip/hip_runtime.h>
#include <stdint.h>

#define BATCH     8
#define NANCH     200000
#define NCLS      20
#define KPRE      500
#define MAXDET    300
#define SCORE_THv 0.05f
#define NMS_THv   0.5f
#define NEGV      (-1e30f)
#define NBINS     4096
#define CHUNK     (NBINS / 256)
#define CAP       1024
#define PITCH     512
#define KW        (PITCH / 32)

typedef unsigned int v4u __attribute__((ext_vector_type(4)));
typedef int          v8i __attribute__((ext_vector_type(8)));
typedef int          v4i __attribute__((ext_vector_type(4)));

__device__ __forceinline__ int bin_of(float s, float lo, float invw) {
    int b = (int)((s - lo) * invw);
    if (b < 0) b = 0;
    if (b > NBINS - 1) b = NBINS - 1;
    return b;
}

// Find the largest bin cb with suffix-count(bins >= cb) >= target.
// Caller pre-inits *outBin = 0, *outAbove = 0 and syncs. 256 threads required.
__device__ void find_cut_bin(const unsigned* hist, unsigned* chunkSum,
                             unsigned* suffixAbove, int* outBin,
                             unsigned* outAbove, unsigned target, int tid) {
    unsigned cs = 0;
#pragma unroll
    for (int i = 0; i < CHUNK; ++i) cs += hist[tid * CHUNK + i];
    chunkSum[tid] = cs;
    __syncthreads();
    if (tid == 0) {
        unsigned run = 0;
        for (int j = 255; j >= 0; --j) { suffixAbove[j] = run; run += chunkSum[j]; }
    }
    __syncthreads();
    unsigned above = suffixAbove[tid];
    if (above < target && above + chunkSum[tid] >= target) {
        unsigned run = above;
        for (int i = CHUNK - 1; i >= 0; --i) {
            unsigned h = hist[tid * CHUNK + i];
            run += h;
            if (run >= target) { *outBin = tid * CHUNK + i; *outAbove = run - h; break; }
        }
    }
    __syncthreads();
}

// Bitonic sort of CAP entries, descending by score, ties -> smaller id first.
__device__ void bitonic_sort_cap(float* s, int* id, int tid, int nthr) {
    for (int k = 2; k <= CAP; k <<= 1) {
        for (int j = k >> 1; j > 0; j >>= 1) {
            __syncthreads();
            for (int i = tid; i < CAP; i += nthr) {
                int p = i ^ j;
                if (p > i) {
                    float si = s[i], sp = s[p];
                    int   ii = id[i], ip = id[p];
                    bool before = (si > sp) || (si == sp && ii < ip);
                    bool desc = ((i & k) == 0);
                    if (before != desc) { s[i] = sp; s[p] = si; id[i] = ip; id[p] = ii; }
                }
            }
        }
    }
    __syncthreads();
}

// ---------------- Phase 1: per-(image,class) exact top-KPRE ----------------
__global__ __launch_bounds__(256) void k_topk_class(const float* __restrict__ cls,
                                                    float* __restrict__ wsS,
                                                    int* __restrict__ wsI) {
    __shared__ unsigned hist[NBINS];
    __shared__ unsigned chunkSum[256];
    __shared__ unsigned suffixAbove[256];
    __shared__ float    cand_s[CAP];
    __shared__ int      cand_i[CAP];
    __shared__ int      cutBin;
    __shared__ unsigned aboveCnt;
    __shared__ unsigned cnt;

    const int bc = blockIdx.x;
    const int b = bc / NCLS, c = bc % NCLS;
    const int tid = threadIdx.x;
    const float* __restrict__ col = cls + (size_t)b * NANCH * NCLS + c;

    for (int i = tid; i < NBINS; i += 256) hist[i] = 0u;
    if (tid == 0) { cutBin = 0; aboveCnt = 0u; cnt = 0u; }
    __syncthreads();

    // Level-1 histogram over [0,1): one DRAM pass; later passes hit the 192MB L2.
    for (int i = tid; i < NANCH; i += 256) {
        if (i + 2048 < NANCH)
            __builtin_prefetch(&col[(size_t)(i + 2048) * NCLS], 0, 3);  // global_prefetch_b8
        float s = col[(size_t)i * NCLS];
        if (s > SCORE_THv) atomicAdd(&hist[bin_of(s, 0.0f, (float)NBINS)], 1u);
    }
    __syncthreads();
    find_cut_bin(hist, chunkSum, suffixAbove, &cutBin, &aboveCnt, KPRE, tid);

    const float lo1 = (float)cutBin * (1.0f / NBINS);
    const float hi1 = lo1 + (1.0f / NBINS);
    const unsigned target2 = KPRE - aboveCnt;
    __syncthreads();               // all threads done reading cutBin/aboveCnt

    for (int i = tid; i < NBINS; i += 256) hist[i] = 0u;
    if (tid == 0) cutBin = 0;
    __syncthreads();

    // Level-2 refinement within the crossing bin (bin width ~6e-8 ≈ 1 ULP near 1.0).
    const float invw2 = (float)NBINS * (float)NBINS;
    for (int i = tid; i < NANCH; i += 256) {
        float s = col[(size_t)i * NCLS];
        if (s > SCORE_THv && s >= lo1 && s < hi1)
            atomicAdd(&hist[bin_of(s, lo1, invw2)], 1u);
    }
    __syncthreads();
    find_cut_bin(hist, chunkSum, suffixAbove, &cutBin, &aboveCnt, target2, tid);
    const float cutVal = lo1 + (float)cutBin * (1.0f / ((float)NBINS * (float)NBINS));

    // Collect candidates (<= ~KPRE + few, far below CAP).
    for (int i = tid; i < NANCH; i += 256) {
        float s = col[(size_t)i * NCLS];
        if (s > SCORE_THv && s >= cutVal) {
            unsigned p = atomicAdd(&cnt, 1u);
            if (p < CAP) { cand_s[p] = s; cand_i[p] = i; }
        }
    }
    __syncthreads();
    unsigned tot = cnt; if (tot > CAP) tot = CAP;
    for (int i = tid; i < CAP; i += 256)
        if (i >= (int)tot) { cand_s[i] = NEGV; cand_i[i] = 0x7FFFFFFF; }
    bitonic_sort_cap(cand_s, cand_i, tid, 256);

    for (int k = tid; k < PITCH; k += 256) {
        float s = (k < KPRE) ? cand_s[k] : NEGV;
        int   id = (k < KPRE) ? cand_i[k] : 0;
        if (id == 0x7FFFFFFF) id = 0;
        wsS[(size_t)bc * PITCH + k] = s;
        wsI[(size_t)bc * PITCH + k] = id;
    }
}

// ---------------- Phase 2: greedy NMS via suppression bitmask ----------------
__global__ __launch_bounds__(512) void k_nms(const float* __restrict__ boxes,
                                             float* __restrict__ wsS,
                                             const int* __restrict__ wsI) {
    __shared__ float    raw[PITCH * 4];          // async-gathered boxes (AoS)
    __shared__ float    X1[PITCH], Y1[PITCH], X2[PITCH], Y2[PITCH], AR[PITCH];
    __shared__ unsigned sup[KPRE * KW];          // 500 x 512-bit suppression matrix
    __shared__ unsigned keep[KW];
    __shared__ unsigned wpre[KW];

    const int bc = blockIdx.x, b = bc / NCLS;
    const int tid = threadIdx.x;

    float sc  = wsS[(size_t)bc * PITCH + tid];
    int   idx = wsI[(size_t)bc * PITCH + tid];
    if (idx < 0) idx = 0;
    if (idx >= NANCH) idx = NANCH - 1;
    const bool valid = (tid < KPRE) && (sc > SCORE_THv);

    // CDNA5 async gather: 16B/lane from global boxes straight into LDS (ASYNCcnt).
    {
        unsigned long long src =
            (unsigned long long)(boxes + ((size_t)b * NANCH + idx) * 4);
        unsigned ldsAddr = (unsigned)(uintptr_t)&raw[tid * 4];
        asm volatile("global_load_async_to_lds_b128 %0, %1, off"
                     :: "v"(ldsAddr), "v"(src) : "memory");
    }
    if (tid < KW) keep[tid] = 0u;
    asm volatile("s_wait_asynccnt 0" ::: "memory");
    __syncthreads();

    {
        float x1 = raw[tid * 4 + 0], y1 = raw[tid * 4 + 1];
        float x2 = raw[tid * 4 + 2], y2 = raw[tid * 4 + 3];
        X1[tid] = x1; Y1[tid] = y1; X2[tid] = x2; Y2[tid] = y2;
        AR[tid] = fmaxf(x2 - x1, 0.0f) * fmaxf(y2 - y1, 0.0f);
    }
    if (valid) atomicOr(&keep[tid >> 5], 1u << (tid & 31));
    __syncthreads();

    // Build sup[i][w]: bit j set iff IoU(i, j) > TH and j > i. Fully parallel.
    for (int t = tid; t < KPRE * KW; t += 512) {
        int i = t >> 4;            // t / KW
        int w = t & (KW - 1);
        float xi1 = X1[i], yi1 = Y1[i], xi2 = X2[i], yi2 = Y2[i], ai = AR[i];
        unsigned m = 0u;
        int jb = w << 5;
        for (int bit = 0; bit < 32; ++bit) {
            int j = jb + bit;
            if (j > i && j < KPRE) {
                float iw = fminf(xi2, X2[j]) - fmaxf(xi1, X1[j]); iw = fmaxf(iw, 0.0f);
                float ih = fminf(yi2, Y2[j]) - fmaxf(yi1, Y1[j]); ih = fmaxf(ih, 0.0f);
                float inter = iw * ih;
                float iou = inter / fmaxf(ai + AR[j] - inter, 1e-8f);
                if (iou > NMS_THv) m |= (1u << bit);
            }
        }
        sup[t] = m;
    }
    __syncthreads();

    // Serial greedy pass by ONE wave32: 512 keep-bits across 16 lanes, no barriers.
    if (tid < 32) {
        int lane = tid;
        unsigned kw = (lane < KW) ? keep[lane] : 0u;
        for (int i = 0; i < KPRE; ++i) {
            unsigned wi = __shfl(kw, i >> 5);
            if ((wi >> (i & 31)) & 1u)
                kw &= ~sup[i * KW + (lane & (KW - 1))];
        }
        if (lane < KW) keep[lane] = kw;
    }
    __syncthreads();

    if (tid == 0) {
        unsigned r = 0;
        for (int w = 0; w < KW; ++w) { wpre[w] = r; r += __popc(keep[w]); }
    }
    __syncthreads();
    {
        unsigned word = keep[tid >> 5];
        bool bit = (word >> (tid & 31)) & 1u;
        unsigned rank = wpre[tid >> 5] + __popc(word & ((1u << (tid & 31)) - 1u));
        bool sel = bit && (rank < MAXDET);
        wsS[(size_t)bc * PITCH + tid] = sel ? sc : NEGV;
    }
}

// ---------------- Phase 3: per-image top-MAXDET merge + emit ----------------
__global__ __launch_bounds__(256) void k_final(const float* __restrict__ boxes,
                                               const float* __restrict__ cls,
                                               const float* __restrict__ wsS,
                                               const int* __restrict__ wsI,
                                               float* __restrict__ outBoxes,
                                               float* __restrict__ outScores,
                                               int* __restrict__ outLabels) {
    __shared__ float    ldsScores[NCLS * KPRE];   // 10000 f32, TDM-packed [c][k]
    __shared__ unsigned hist[NBINS];              // reused as cand_s/cand_f later
    __shared__ unsigned chunkSum[256];
    __shared__ unsigned suffixAbove[256];
    __shared__ int      cutBin;
    __shared__ unsigned aboveCnt;
    __shared__ unsigned cnt;

    const int b = blockIdx.x, tid = threadIdx.x;
    const float* wsb = wsS + (size_t)b * NCLS * PITCH;

    if (tid == 0) { cutBin = 0; aboveCnt = 0u; cnt = 0u; }
    for (int i = tid; i < NBINS; i += 256) hist[i] = 0u;

    // CDNA5 TDM: one 2D descriptor DMAs the 20x500 score tile (row pitch 512)
    // into contiguous LDS; LDS index == reference flat index f = c*500 + k.
    // Issued by wave 0 only (TDM ignores EXEC; avoids duplicate DMAs).
    // clang-23 lane: 6-arg builtin (g0 v4u, g1 v8i, g2 v4i, g3 v4i, v8i, cpol).
    if (tid < 32) {
        unsigned long long ga = (unsigned long long)wsb;
        unsigned ldsA = (unsigned)(uintptr_t)&ldsScores[0];
        v4u g0 = { 1u,                                     // count=1 (valid user D#)
                   ldsA,                                   // lds_addr
                   (unsigned)(ga & 0xFFFFFFFFull),         // global_addr[31:0]
                   (unsigned)((ga >> 32) & 0x1FFFFFFull) | (2u << 30) }; // [56:32]|type=2
        v8i g1 = { (int)(2u << 16),                        // data_size=4B
                   (int)((unsigned)PITCH << 16),           // tensor_dim0 = 512
                   (int)((unsigned)NCLS << 16),            // tensor_dim1 = 20
                   (int)((unsigned)KPRE << 16),            // tile_dim0  = 500
                   (int)(unsigned)NCLS,                    // tile_dim1  = 20
                   (int)(unsigned)PITCH,                   // tensor_dim0_stride = 512
                   0, 0 };
        v4i gz  = { 0, 0, 0, 0 };                          // 2D tensor: groups 2/3 unused
        v8i gz8 = { 0, 0, 0, 0, 0, 0, 0, 0 };              // extra group: NULL (count=0)
        __builtin_amdgcn_tensor_load_to_lds(g0, g1, gz, gz, gz8, 0);
        __builtin_amdgcn_s_wait_tensorcnt(0);
    }
    __syncthreads();

    for (int f = tid; f < NCLS * KPRE; f += 256) {
        float s = ldsScores[f];
        if (s > SCORE_THv) atomicAdd(&hist[bin_of(s, 0.0f, (float)NBINS)], 1u);
    }
    __syncthreads();
    find_cut_bin(hist, chunkSum, suffixAbove, &cutBin, &aboveCnt, MAXDET, tid);

    const float lo1 = (float)cutBin * (1.0f / NBINS);
    const float hi1 = lo1 + (1.0f / NBINS);
    const unsigned target2 = MAXDET - aboveCnt;
    __syncthreads();

    for (int i = tid; i < NBINS; i += 256) hist[i] = 0u;
    if (tid == 0) cutBin = 0;
    __syncthreads();
    const float invw2 = (float)NBINS * (float)NBINS;
    for (int f = tid; f < NCLS * KPRE; f += 256) {
        float s = ldsScores[f];
        if (s > SCORE_THv && s >= lo1 && s < hi1)
            atomicAdd(&hist[bin_of(s, lo1, invw2)], 1u);
    }
    __syncthreads();
    find_cut_bin(hist, chunkSum, suffixAbove, &cutBin, &aboveCnt, target2, tid);
    const float cutVal = lo1 + (float)cutBin * (1.0f / ((float)NBINS * (float)NBINS));

    // Histogram is dead from here on: overlay candidate buffers on it (saves 8KB LDS).
    float* cand_s = (float*)hist;
    int*   cand_f = (int*)(hist + CAP);

    for (int f = tid; f < NCLS * KPRE; f += 256) {
        float s = ldsScores[f];
        if (s > SCORE_THv && s >= cutVal) {
            unsigned p = atomicAdd(&cnt, 1u);
            if (p < CAP) { cand_s[p] = s; cand_f[p] = f; }
        }
    }
    __syncthreads();
    unsigned tot = cnt; if (tot > CAP) tot = CAP;
    for (int i = tid; i < CAP; i += 256)
        if (i >= (int)tot) { cand_s[i] = NEGV; cand_f[i] = 0x7FFFFFFF; }
    bitonic_sort_cap(cand_s, cand_f, tid, 256);

    for (int k = tid; k < MAXDET; k += 256) {
        float v = cand_s[k];
        bool valid = v > SCORE_THv;
        int f = valid ? cand_f[k] : 0;
        int c = f / KPRE;
        int kk = f - c * KPRE;
        int fa = valid ? wsI[((size_t)b * NCLS + c) * PITCH + kk] : 0;
        if (fa < 0) fa = 0;
        if (fa >= NANCH) fa = NANCH - 1;
        const float* bp = boxes + ((size_t)b * NANCH + fa) * 4;
        size_t o = (size_t)b * MAXDET + k;
        outBoxes[o * 4 + 0] = valid ? bp[0] : -1.0f;
        outBoxes[o * 4 + 1] = valid ? bp[1] : -1.0f;
        outBoxes[o * 4 + 2] = valid ? bp[2] : -1.0f;
        outBoxes[o * 4 + 3] = valid ? bp[3] : -1.0f;
        outScores[o] = valid ? v : -1.0f;       // v == cls[b, fa, c] exactly
        outLabels[o] = valid ? (c - 1) : -1;    // reference emits label-1
    }
}

extern "C" void kernel_launch(void* const* d_in, const int* in_sizes, int n_in,
                              void* d_out, int out_size, void* d_ws, size_t ws_size,
                              hipStream_t stream) {
    const float* boxes = (const float*)d_in[0];
    const float* cls   = (const float*)d_in[1];

    float* out       = (float*)d_out;
    float* outBoxes  = out;                                      // [B,300,4]
    float* outScores = out + (size_t)BATCH * MAXDET * 4;         // [B,300]
    int*   outLabels = (int*)(out + (size_t)BATCH * MAXDET * 5); // [B,300] i32

    float* wsS = (float*)d_ws;                                   // [B*C, 512] f32
    int*   wsI = (int*)((char*)d_ws +
                        (size_t)BATCH * NCLS * PITCH * sizeof(float));

    k_topk_class<<<BATCH * NCLS, 256, 0, stream>>>(cls, wsS, wsI);
    k_nms<<<BATCH * NCLS, PITCH, 0, stream>>>(boxes, wsS, wsI);
    k_final<<<BATCH, 256, 0, stream>>>(boxes, cls, wsS, wsI,
                                       outBoxes, outScores, outLabels);
}